// RNNClass_23596550324253
// MI455X (gfx1250) — compile-verified
//
#include <hip/hip_runtime.h>
#include <hip/hip_bf16.h>

typedef __attribute__((ext_vector_type(2))) float v2f;
typedef __attribute__((ext_vector_type(4))) float v4f;
typedef __attribute__((ext_vector_type(8))) float v8f;

#define ACTION_COST 0.1f

// ---------------------------------------------------------------------------
// Kernel 1: prefix-scan the 1024 actions, emit per-timestep B-vector
//   bvec[t] = { 1, cx[t], cy[t], r[t] }  with
//   cx,cy = inclusive cumsum of actions, r = -(cx^2+cy^2) - c*(ax^2+ay^2)
// One block, blockDim.x == horizon (1024). Hillis-Steele scan in LDS.
// ---------------------------------------------------------------------------
__global__ void rnn_prep_kernel(const float* __restrict__ actions,
                                v4f* __restrict__ bvec, int horizon) {
    __shared__ float2 s[1024];
    const int t = threadIdx.x;
    float ax = actions[2 * t + 0];
    float ay = actions[2 * t + 1];
    s[t] = make_float2(ax, ay);
    __syncthreads();
    for (int off = 1; off < horizon; off <<= 1) {
        float2 v = s[t];
        float2 u = make_float2(0.0f, 0.0f);
        if (t >= off) u = s[t - off];
        __syncthreads();
        v.x += u.x;
        v.y += u.y;
        s[t] = v;
        __syncthreads();
    }
    float cx = s[t].x;
    float cy = s[t].y;
    float r  = -(cx * cx + cy * cy) - ACTION_COST * (ax * ax + ay * ay);
    v4f b;
    b.x = 1.0f; b.y = cx; b.z = cy; b.w = r;
    bvec[t] = b;
}

// ---------------------------------------------------------------------------
// Kernel 2: rank-4 outer-product expansion as K=4 f32 WMMA.
//   reward[t][b] = p[b]*1 + qx[b]*cx[t] + qy[b]*cy[t] + 1*r[t]
// One wave handles a 16-batch tile (A matrix, built once) and loops over
// 16 time-tiles of 16 timesteps each. Each v_wmma_f32_16x16x4_f32 produces
// a 16x16 f32 tile; each lane stores its 8 contiguous results as two
// non-temporal 128-bit stores (output >> L2, so stream past cache).
//
// VGPR layouts (CDNA5 ISA §7.12.2, 32-bit matrices):
//   A 16x4 : lanes 0-15 -> {K0,K1}, lanes 16-31 -> {K2,K3}, M = lane%16
//   B 4x16 : lanes 0-15 -> {K0,K1}, lanes 16-31 -> {K2,K3}, N = lane%16
//   D 16x16: VGPR v -> M = v + 8*(lane/16), N = lane%16
// ---------------------------------------------------------------------------
__global__ void rnn_wmma_kernel(const float* __restrict__ init_state,
                                const float* __restrict__ goal,
                                const v4f* __restrict__ bvec,
                                float* __restrict__ out, int batch) {
    const int lane = threadIdx.x & 31;
    const int wave = threadIdx.x >> 5;      // 0..7
    const int half = lane >> 4;             // 0: K=0,1 / M=0..7 ; 1: K=2,3 / M=8..15
    const int idx  = lane & 15;             // M (batch row) for A, N (time col) for B/D

    // ---- Build A row for this wave's 16 batch elements ----
    const int b0 = (blockIdx.x * 8 + wave) * 16;
    const float gx = goal[0];
    const float gy = goal[1];
    const float2 st = ((const float2*)init_state)[b0 + idx];
    const float dx = st.x - gx;
    const float dy = st.y - gy;
    const float p  = -(dx * dx + dy * dy);
    const float qx = -2.0f * dx;
    const float qy = -2.0f * dy;
    v2f a;                                   // A row = (p, qx, qy, 1)
    a.x = half ? qy   : p;
    a.y = half ? 1.0f : qx;

    // ---- Loop over 16 time-tiles (16 timesteps each) ----
    const int ttbase = blockIdx.y * 16;
#pragma unroll 4
    for (int tt = 0; tt < 16; ++tt) {
        const int t0 = (ttbase + tt) * 16;
        const v4f bv = bvec[t0 + idx];       // {1, cx, cy, r} for timestep t0+idx
        v2f b;                               // B col = (1, cx, cy, r)
        b.x = half ? bv.z : bv.x;
        b.y = half ? bv.w : bv.y;

        v8f c = {};
        // D = A(16x4) x B(4x16) + 0 : emits v_wmma_f32_16x16x4_f32
        v8f d = __builtin_amdgcn_wmma_f32_16x16x4_f32(
            /*neg_a=*/false, a, /*neg_b=*/false, b,
            /*c_mod=*/(short)0, c, /*reuse_a=*/false, /*reuse_b=*/false);

        // Lane's 8 results: column t = t0+idx, rows b = b0 + 8*half + v (contiguous)
        float* pout = out + (size_t)(t0 + idx) * (size_t)batch + b0 + 8 * half;
        v4f lo; lo.x = d[0]; lo.y = d[1]; lo.z = d[2]; lo.w = d[3];
        v4f hi; hi.x = d[4]; hi.y = d[5]; hi.z = d[6]; hi.w = d[7];
        __builtin_nontemporal_store(lo, (v4f*)pout);
        __builtin_nontemporal_store(hi, (v4f*)(pout + 4));
    }
}

extern "C" void kernel_launch(void* const* d_in, const int* in_sizes, int n_in,
                              void* d_out, int out_size, void* d_ws, size_t ws_size,
                              hipStream_t stream) {
    const float* actions    = (const float*)d_in[0];  // (horizon, 2)
    const float* init_state = (const float*)d_in[1];  // (batch, 2)
    const float* goal       = (const float*)d_in[2];  // (2,)
    float* out = (float*)d_out;                       // (horizon, batch)

    const int horizon = in_sizes[0] / 2;              // 1024
    const int batch   = in_sizes[1] / 2;              // 65536

    v4f* bvec = (v4f*)d_ws;                           // horizon * 16 bytes

    // Stage 1: scan actions -> per-timestep B vectors
    rnn_prep_kernel<<<1, horizon, 0, stream>>>(actions, bvec, horizon);

    // Stage 2: WMMA outer-product expansion + streaming stores
    // grid.x: groups of 8 batch-tiles (one per wave), grid.y: chunks of 16 time-tiles
    dim3 grid(batch / (16 * 8), horizon / (16 * 16));
    rnn_wmma_kernel<<<grid, 256, 0, stream>>>(init_state, goal, bvec, out, batch);
}